// Kernel_5552027089989569678_77051713290577
// MI455X (gfx1250) — compile-verified
//
#include <hip/hip_runtime.h>
#include <hip/hip_bf16.h>
#include <math.h>

typedef __attribute__((ext_vector_type(2))) float v2f;
typedef __attribute__((ext_vector_type(8))) float v8f;

#define NB 16
#define CC 192
#define HH 56
#define WW 56
#define SS (HH*WW)        // 3136
#define KIN 672           // input channels per group for t6 GEMM
#define CH 96             // output channels per group

#define WMMA_F32(A,B,C) __builtin_amdgcn_wmma_f32_16x16x4_f32(false,(A),false,(B),(short)0,(C),false,false)

// -------- kernel 1: depthwise 5x5, pad 2 --------
__global__ void k_dw5(const float* __restrict__ x, const float* __restrict__ w1,
                      float* __restrict__ t1) {
  int idx = blockIdx.x * blockDim.x + threadIdx.x;
  const int total = NB*CC*SS;
  if (idx >= total) return;
  int s  = idx % SS;
  int nc = idx / SS;
  int c  = nc % CC;
  int y  = s / WW, xx = s % WW;
  const float* wp = w1 + c*25;
  const float* xp = x + (size_t)nc*SS;
  float acc = 0.f;
#pragma unroll
  for (int ky = 0; ky < 5; ++ky) {
    int yy = y + ky - 2;
    if (yy < 0 || yy >= HH) continue;
#pragma unroll
    for (int kx = 0; kx < 5; ++kx) {
      int xs = xx + kx - 2;
      if (xs < 0 || xs >= WW) continue;
      acc = fmaf(wp[ky*5+kx], xp[yy*WW+xs], acc);
    }
  }
  t1[idx] = acc;
}

// -------- kernel 2: gram t2[n] = X Xt / 56, 2x2 blocked, upper-triangle only --------
// t2 is symmetric: compute 21 of 36 (32x32) blocks, store block + mirror.
__global__ void k_gram_wmma(const float* __restrict__ x, float* __restrict__ t2) {
  int wave = (blockIdx.x * blockDim.x + threadIdx.x) >> 5;
  int lane = threadIdx.x & 31;
  int tri = wave % 21;
  int n   = wave / 21;
  if (n >= NB) return;
  // triangular decode: (it2 <= jt2) among 6x6 blocks
  int it2 = 0, rem = tri;
  while (rem >= 6 - it2) { rem -= 6 - it2; ++it2; }
  int jt2 = it2 + rem;

  int l15 = lane & 15, lhalf = lane >> 4;
  int koff = lhalf * 2;                     // K pair per half-wave (ISA A/B layout)
  const float* Xn = x + (size_t)n*CC*SS;
  const float* pa0 = Xn + (size_t)(it2*32 + l15     )*SS + koff;
  const float* pa1 = Xn + (size_t)(it2*32 + 16 + l15)*SS + koff;
  const float* pb0 = Xn + (size_t)(jt2*32 + l15     )*SS + koff;
  const float* pb1 = Xn + (size_t)(jt2*32 + 16 + l15)*SS + koff;
  v8f c00 = {}, c01 = {}, c10 = {}, c11 = {};
  for (int k = 0; k < SS; k += 4) {
    v2f a0 = *(const v2f*)(pa0 + k);
    v2f a1 = *(const v2f*)(pa1 + k);
    v2f b0 = *(const v2f*)(pb0 + k);
    v2f b1 = *(const v2f*)(pb1 + k);
    c00 = WMMA_F32(a0, b0, c00);
    c01 = WMMA_F32(a0, b1, c01);
    c10 = WMMA_F32(a1, b0, c10);
    c11 = WMMA_F32(a1, b1, c11);
  }
  const float sc = 1.0f / 56.0f;            // 1/sqrt(3136)
  float* o = t2 + (size_t)n*CC*CC;
  int rbase = lhalf * 8;
#pragma unroll
  for (int j = 0; j < 8; ++j) {
    int r0 = it2*32 + rbase + j, r1 = r0 + 16;
    int q0 = jt2*32 + l15,       q1 = q0 + 16;
    float v00 = c00[j]*sc, v01 = c01[j]*sc, v10 = c10[j]*sc, v11 = c11[j]*sc;
    o[(size_t)r0*CC + q0] = v00;  o[(size_t)q0*CC + r0] = v00;
    o[(size_t)r0*CC + q1] = v01;  o[(size_t)q1*CC + r0] = v01;
    o[(size_t)r1*CC + q0] = v10;  o[(size_t)q0*CC + r1] = v10;
    o[(size_t)r1*CC + q1] = v11;  o[(size_t)q1*CC + r1] = v11;
  }
}

// -------- kernel 3: fused t5 construction + grouped 1x1 GEMM + GELU --------
// One wave owns the full M=96 (6 tiles) for one 16-pixel column tile:
// B fragment is built once per k-step and fed to 6 WMMAs.
__global__ void k_t6_wmma(const float* __restrict__ x, const float* __restrict__ t1,
                          const float* __restrict__ w5, const float* __restrict__ w6,
                          float* __restrict__ t7) {
  int wave = (blockIdx.x * blockDim.x + threadIdx.x) >> 5;
  int lane = threadIdx.x & 31;
  const int NT = (NB*SS) / 16;              // 3136 pixel tiles
  int nt = wave % NT;
  int g  = wave / NT;
  if (g >= 2) return;

  int l15 = lane & 15, lhalf = lane >> 4;
  int koff = lhalf * 2;

  // A fragments: 6 weight rows (one per 16-row M tile), contiguous K pairs
  const float* wrow = w6 + (size_t)(g*CH + l15)*KIN + koff;   // + mtile*16*KIN

  // B column: global pixel -> (n, s, y, x)
  int p   = nt*16 + l15;
  int n   = p / SS;
  int s   = p % SS;
  int y   = s / WW;
  int xxp = s % WW;
  int gbase = g*CH;                         // input channel-group base
  const float* xcol  = x  + ((size_t)n*CC + gbase)*SS + s;
  const float* t1col = t1 + ((size_t)n*CC + gbase)*SS + xxp;
  const float* w5p   = w5 + gbase*7;

  v8f acc[6] = {{}, {}, {}, {}, {}, {}};
  for (int k = 0; k < KIN; k += 4) {
    // ---- build B fragment on the fly (L2-resident gathers) ----
    int ci0 = k + koff;
    float be[2];
#pragma unroll
    for (int e = 0; e < 2; ++e) {
      int ci = ci0 + e;
      int cq = ci / 7;                      // channel within group
      int ct = ci - cq*7;                   // tap
      float xv = xcol[(size_t)cq*SS];
      int yy = y + 3*ct - 9;
      float tv = (yy >= 0 && yy < HH) ? t1col[(size_t)cq*SS + yy*WW] : 0.f;
      be[e] = w5p[cq*7 + ct] * (xv + tv);
    }
    v2f b = { be[0], be[1] };
    // ---- 6 WMMAs reuse this B ----
#pragma unroll
    for (int m = 0; m < 6; ++m) {
      v2f a = *(const v2f*)(wrow + (size_t)m*16*KIN + k);
      acc[m] = WMMA_F32(a, b, acc[m]);
    }
  }
  int rbase = lhalf * 8;
#pragma unroll
  for (int m = 0; m < 6; ++m) {
#pragma unroll
    for (int j = 0; j < 8; ++j) {
      int coo = g*CH + m*16 + rbase + j;
      float v  = acc[m][j];
      float ge = 0.5f * v * (1.0f + erff(v * 0.70710678118654752440f));
      t7[((size_t)n*CC + coo)*SS + s] = ge;
    }
  }
}

// -------- kernel 4: depthwise 7x7, dilation 3, pad 9; TRANSPOSED output --------
// Writes t8T[n][s][c] so the t9 GEMM's B fragments are contiguous v2f loads.
__global__ void k_dw7d3(const float* __restrict__ t7, const float* __restrict__ w8,
                        float* __restrict__ t8T) {
  int idx = blockIdx.x * blockDim.x + threadIdx.x;
  const int total = NB*CC*SS;
  if (idx >= total) return;
  int s  = idx % SS;
  int nc = idx / SS;
  int c  = nc % CC;
  int n  = nc / CC;
  int y  = s / WW, xx = s % WW;
  const float* wp = w8 + c*49;
  const float* tp = t7 + (size_t)nc*SS;
  float acc = 0.f;
#pragma unroll
  for (int ky = 0; ky < 7; ++ky) {
    int yy = y + 3*ky - 9;
    if (yy < 0 || yy >= HH) continue;
#pragma unroll
    for (int kx = 0; kx < 7; ++kx) {
      int xs = xx + 3*kx - 9;
      if (xs < 0 || xs >= WW) continue;
      acc = fmaf(wp[ky*7+kx], tp[yy*WW+xs], acc);
    }
  }
  t8T[((size_t)n*SS + s)*CC + c] = acc;     // scattered store; reads stay coalesced
}

// -------- kernel 5: t9[n] = t2[n]^T * t8[n] / sqrt(192), 2x2 blocked --------
// t2 symmetric => A = t2 rows (contiguous). t8 pre-transposed => B contiguous.
__global__ void k_t9_wmma(const float* __restrict__ t2, const float* __restrict__ t8T,
                          float* __restrict__ out) {
  int wave = (blockIdx.x * blockDim.x + threadIdx.x) >> 5;
  int lane = threadIdx.x & 31;
  const int NT2 = SS / 32;                  // 98 column blocks of 32
  int nt2 = wave % NT2;
  int t   = wave / NT2;
  int mt2 = t % 6;                          // 32-row block
  int n   = t / 6;
  if (n >= NB) return;
  int l15 = lane & 15, lhalf = lane >> 4;
  int koff = lhalf * 2;
  const float* t2n = t2  + (size_t)n*CC*CC;
  const float* t8n = t8T + (size_t)n*SS*CC;
  int col0 = nt2*32 + l15, col1 = col0 + 16;
  const float* pa0 = t2n + (size_t)(mt2*32 + l15     )*CC + koff;
  const float* pa1 = t2n + (size_t)(mt2*32 + 16 + l15)*CC + koff;
  const float* pb0 = t8n + (size_t)col0*CC + koff;
  const float* pb1 = t8n + (size_t)col1*CC + koff;
  v8f c00 = {}, c01 = {}, c10 = {}, c11 = {};
  for (int k = 0; k < CC; k += 4) {
    v2f a0 = *(const v2f*)(pa0 + k);
    v2f a1 = *(const v2f*)(pa1 + k);
    v2f b0 = *(const v2f*)(pb0 + k);
    v2f b1 = *(const v2f*)(pb1 + k);
    c00 = WMMA_F32(a0, b0, c00);
    c01 = WMMA_F32(a0, b1, c01);
    c10 = WMMA_F32(a1, b0, c10);
    c11 = WMMA_F32(a1, b1, c11);
  }
  const float sc = 0.07216878364870322135f; // 1/sqrt(192)
  int rbase = lhalf * 8;
#pragma unroll
  for (int j = 0; j < 8; ++j) {
    int r0 = mt2*32 + rbase + j, r1 = r0 + 16;
    out[((size_t)n*CC + r0)*SS + col0] = c00[j] * sc;
    out[((size_t)n*CC + r0)*SS + col1] = c01[j] * sc;
    out[((size_t)n*CC + r1)*SS + col0] = c10[j] * sc;
    out[((size_t)n*CC + r1)*SS + col1] = c11[j] * sc;
  }
}

extern "C" void kernel_launch(void* const* d_in, const int* in_sizes, int n_in,
                              void* d_out, int out_size, void* d_ws, size_t ws_size,
                              hipStream_t stream) {
  const float* x  = (const float*)d_in[0];
  const float* w1 = (const float*)d_in[1];
  const float* w5 = (const float*)d_in[2];
  const float* w6 = (const float*)d_in[3];
  const float* w8 = (const float*)d_in[4];
  float* out = (float*)d_out;

  const size_t NCS = (size_t)NB*CC*SS;
  float* t1  = (float*)d_ws;                   // 38.5 MB
  float* t2  = t1 + NCS;                       // 2.4 MB
  float* t7  = t2 + (size_t)NB*CC*CC;          // 38.5 MB
  float* t8T = t1;                             // reuse t1 region (dead after k_t6)

  const int total = NB*CC*SS;                  // 9,633,792
  k_dw5      <<<(total + 255) / 256, 256, 0, stream>>>(x, w1, t1);
  k_gram_wmma<<<(NB*21) / 8 + 1,     256, 0, stream>>>(x, t2);            // 336 waves
  k_t6_wmma  <<<(2*((NB*SS)/16)) / 8, 256, 0, stream>>>(x, t1, w5, w6, t7); // 6272 waves
  k_dw7d3    <<<(total + 255) / 256, 256, 0, stream>>>(t7, w8, t8T);
  k_t9_wmma  <<<(NB*6*(SS/32)) / 8,  256, 0, stream>>>(t2, t8T, out);     // 9408 waves
}